// Conv1D_35390530519543
// MI455X (gfx1250) — compile-verified
//
#include <hip/hip_runtime.h>
#include <hip/hip_bf16.h>

typedef __attribute__((ext_vector_type(2))) float v2f;
typedef __attribute__((ext_vector_type(8))) float v8f;

#define N_LAYERS 200
#define KSIZE    7
#define L_IN     1388
#define FC_IN    188
#define FC_OUT   91
#define BATCH    1024
#define BT       16     // batch tile = WMMA N
#define RS       20     // LDS row stride (floats): bank-conflict-free vs 64 banks
#define ROWS     1400   // rows per buffer: >= max unguarded read row (1399)+1
#define NW       8      // waves per block (blockDim=256, wave32)
#define HBUF     (ROWS * RS)

__global__ __launch_bounds__(256) void convnet_wmma_kernel(
    const float* __restrict__ x,      // [1024,1388]
    const float* __restrict__ conv_w, // [200,7]
    const float* __restrict__ conv_b, // [200]
    const float* __restrict__ fc_w,   // [91,188]
    const float* __restrict__ fc_b,   // [91]
    float* __restrict__ out)          // [1024,91]
{
  extern __shared__ float lds[];
  float* bufA = lds;
  float* bufB = lds + HBUF;
  float* taps = lds + 2 * HBUF;   // taps[2][8]; slot [.][7] stays 0 forever

  const int tid  = threadIdx.x;
  const int lane = tid & 31;
  const int wave = tid >> 5;
  const int m    = lane & 15;     // A: M row | B,D: N column (batch)
  const int half = lane >> 4;     // lane half selects K pair {0,1} vs {2,3}
  const int kb   = half * 2;

  const int b0 = blockIdx.x * BT;

  // ---- init: zero bufA tail rows [L_IN,ROWS) (only region read-before-written),
  //      stage taps slot 0, zero the two "zero" tap slots, stage x. ----
  for (int i = tid; i < (ROWS - L_IN) * RS; i += 256)
    bufA[L_IN * RS + i] = 0.0f;
  if (tid < KSIZE) taps[tid] = conv_w[tid];
  if (tid == KSIZE) { taps[7] = 0.0f; taps[15] = 0.0f; }
  for (int n = 0; n < BT; ++n) {
    const float* xr = x + (size_t)(b0 + n) * L_IN;
    for (int j = tid; j < L_IN; j += 256)
      bufA[j * RS + n] = xr[j];
  }
  __syncthreads();

  // Layer-invariant banded-A tap indices (7 -> zero slot)
  int tix[6], tiy[6];
#pragma unroll
  for (int s = 0; s < 6; ++s) {
    int t0 = 4 * s + kb - m;
    tix[s] = ((unsigned)t0       <= 6u) ? t0       : 7;
    tiy[s] = ((unsigned)(t0 + 1) <= 6u) ? (t0 + 1) : 7;
  }

  float* hin  = bufA;
  float* hout = bufB;
  int Lin = L_IN;

  // ---- 200 sequential banded-matmul conv layers on fp32 WMMA ----
  for (int layer = 0; layer < N_LAYERS; ++layer) {
    const int Lout = Lin - (KSIZE - 1);
    const float bias = conv_b[layer];
    const float* tp = taps + (layer & 1) * 8;

    v2f afrag[6];
#pragma unroll
    for (int s = 0; s < 6; ++s) {
      afrag[s].x = tp[tix[s]];
      afrag[s].y = tp[tiy[s]];
    }

    const int ntiles = (Lout + BT - 1) >> 4;
    for (int t = wave; t < ntiles; t += NW) {
      const int j0 = t << 4;
      const float* bp = hin + (j0 + kb) * RS + m;   // B base for this lane
      v8f acc;
#pragma unroll
      for (int r = 0; r < 8; ++r) acc[r] = bias;

#pragma unroll
      for (int s = 0; s < 6; ++s) {
        v2f bf;
        bf.x = bp[(4 * s) * RS];          // constant byte offsets: s*320
        bf.y = bp[(4 * s + 1) * RS];      //                       s*320+80
        acc = __builtin_amdgcn_wmma_f32_16x16x4_f32(
            false, afrag[s], false, bf, (short)0, acc, false, false);
      }

      if (layer < N_LAYERS - 1) {
#pragma unroll
        for (int r = 0; r < 8; ++r) acc[r] = fmaxf(acc[r], 0.0f);
      }

      float* sp = hout + (j0 + 8 * half) * RS + m;  // D: VGPR r -> M=r+8*half
#pragma unroll
      for (int r = 0; r < 8; ++r) sp[r * RS] = acc[r];
    }

    // stage taps for next layer into the other slot (race-free: slot was last
    // read before the *previous* barrier)
    if (layer + 1 < N_LAYERS && tid < KSIZE)
      taps[((layer + 1) & 1) * 8 + tid] = conv_w[(layer + 1) * KSIZE + tid];

    __syncthreads();
    float* tmp = hin; hin = hout; hout = tmp;
    Lin = Lout;
  }
  // hin holds h[188][16] in LDS

  // ---- FC 188->91 + sigmoid: M=91 (6 tiles), N=16 batch, K=188=47*4 ----
  for (int ot = wave; ot < 6; ot += NW) {
    const int o0 = ot << 4;
    const int om = o0 + m;
    const bool aval = (om < FC_OUT);
    const float* ap = fc_w + (size_t)(aval ? om : (FC_OUT - 1)) * FC_IN;
    const float* bp = hin + kb * RS + m;

    v8f acc;
#pragma unroll
    for (int r = 0; r < 8; ++r) {
      int o = o0 + r + 8 * half;
      acc[r] = (o < FC_OUT) ? fc_b[o < FC_OUT ? o : 0] : 0.0f;
    }
    for (int s = 0; s < 47; ++s) {
      int k0 = 4 * s;
      v2f a, b;
      a.x = aval ? ap[k0 + kb]     : 0.0f;
      a.y = aval ? ap[k0 + kb + 1] : 0.0f;
      b.x = bp[k0 * RS];
      b.y = bp[(k0 + 1) * RS];
      acc = __builtin_amdgcn_wmma_f32_16x16x4_f32(
          false, a, false, b, (short)0, acc, false, false);
    }
#pragma unroll
    for (int r = 0; r < 8; ++r) {
      int o = o0 + r + 8 * half;
      if (o < FC_OUT) {
        float v = 1.0f / (1.0f + __expf(-acc[r]));
        out[(size_t)(b0 + m) * FC_OUT + o] = v;
      }
    }
  }
}

extern "C" void kernel_launch(void* const* d_in, const int* in_sizes, int n_in,
                              void* d_out, int out_size, void* d_ws, size_t ws_size,
                              hipStream_t stream) {
  (void)in_sizes; (void)n_in; (void)out_size; (void)d_ws; (void)ws_size;
  const float* x      = (const float*)d_in[0];
  const float* conv_w = (const float*)d_in[1];
  const float* conv_b = (const float*)d_in[2];
  const float* fc_w   = (const float*)d_in[3];
  const float* fc_b   = (const float*)d_in[4];
  float* out = (float*)d_out;

  const size_t shmem = (size_t)(2 * HBUF + 16) * sizeof(float); // 224,064 B < 320 KB
  (void)hipFuncSetAttribute((const void*)convnet_wmma_kernel,
                            hipFuncAttributeMaxDynamicSharedMemorySize, (int)shmem);
  convnet_wmma_kernel<<<dim3(BATCH / BT), dim3(256), shmem, stream>>>(
      x, conv_w, conv_b, fc_w, fc_b, out);
}